// LNon_9320079033245
// MI455X (gfx1250) — compile-verified
//
#include <hip/hip_runtime.h>
#include <hip/hip_bf16.h>
#include <math.h>

typedef float v2f __attribute__((ext_vector_type(2)));
typedef float v8f __attribute__((ext_vector_type(8)));

#define NBLK_RED 1024
#define TPB_RED  256

// ---- workspace layout (float offsets) ----
#define WS_STATS 0      // [0]=mean1 [1]=invstd1*iscale [2]=mean2 [3]=invstd2*oscale
#define WS_THMOD 16     // 16 floats: theta[p]*rowsum[p]
#define WS_VMOD  32     // 16 floats: velocity[p]*rowsum[p]
#define WS_P1    1024   // 2*NBLK_RED partial (sum,sumsq) for input stats
#define WS_P2    3072   // 2*NBLK_RED partial (sum,sumsq) for output stats
#define WS_T1    5120   // 256*16  : T1 = x0 (256x1024) * sp1 (1024x16pad)
#define WS_T2    9216   // 1024*16 : T2 = x0^T (1024x256) * ch2 (256x16pad)

// ---------------- pass 1: input stats (deterministic two-stage) -------------
__global__ void k_stats1(const float* __restrict__ d, long long n, float* ws) {
    __shared__ float ssum[TPB_RED], ssq[TPB_RED];
    float s = 0.f, q = 0.f;
    long long stride = (long long)gridDim.x * blockDim.x;
    for (long long i = (long long)blockIdx.x * blockDim.x + threadIdx.x; i < n; i += stride) {
        float v = d[i]; s += v; q += v * v;
    }
    ssum[threadIdx.x] = s; ssq[threadIdx.x] = q;
    __syncthreads();
    for (int o = TPB_RED >> 1; o > 0; o >>= 1) {
        if ((int)threadIdx.x < o) {
            ssum[threadIdx.x] += ssum[threadIdx.x + o];
            ssq [threadIdx.x] += ssq [threadIdx.x + o];
        }
        __syncthreads();
    }
    if (threadIdx.x == 0) {
        ws[WS_P1 + 2 * blockIdx.x]     = ssum[0];
        ws[WS_P1 + 2 * blockIdx.x + 1] = ssq[0];
    }
}

__global__ void k_stats1b(float* ws, long long n, const float* __restrict__ iscale) {
    __shared__ float ssum[NBLK_RED], ssq[NBLK_RED];
    int t = threadIdx.x;
    ssum[t] = ws[WS_P1 + 2 * t];
    ssq [t] = ws[WS_P1 + 2 * t + 1];
    __syncthreads();
    for (int o = NBLK_RED >> 1; o > 0; o >>= 1) {
        if (t < o) { ssum[t] += ssum[t + o]; ssq[t] += ssq[t + o]; }
        __syncthreads();
    }
    if (t == 0) {
        double dn = (double)n;
        double m  = (double)ssum[0] / dn;
        double var = ((double)ssq[0] - dn * m * m) / (dn - 1.0);
        float inv = (float)(1.0 / sqrt(var));
        ws[0] = (float)m;
        ws[1] = inv * iscale[0];   // fold iscale: x = (d - mean) * ws[1]
    }
}

// ---------------- pass 2: weight GEMMs via V_WMMA_F32_16X16X4_F32 -----------
// A-frag layout (16x4 f32): lanes 0-15 hold (K0,K1), lanes 16-31 hold (K2,K3)
// for M = lane&15. B-frag mirrors with N = lane&15. D: vgpr r -> row r(+8), col lane.

// T1[c,p] = sum_s x0[c,s]*sp1[s,p];  one wave per 16-row tile of c.
__global__ void k_t1(const float* __restrict__ data, const float* __restrict__ sp1,
                     float* ws) {
    const float mean = ws[0], sc = ws[1];
    const int lane = threadIdx.x, half = lane >> 4, l = lane & 15;
    const int c0 = blockIdx.x * 16;
    const int lc = (l < 12) ? l : 0;
    v8f acc = {};
    for (int s = 0; s < 1024; s += 4) {
        const int k0 = s + 2 * half;
        v2f a, b;
        a.x = (data[(c0 + l) * 1024 + k0]     - mean) * sc;
        a.y = (data[(c0 + l) * 1024 + k0 + 1] - mean) * sc;
        float bx = sp1[k0 * 12 + lc];
        float by = sp1[(k0 + 1) * 12 + lc];
        b.x = (l < 12) ? bx : 0.f;
        b.y = (l < 12) ? by : 0.f;
        acc = __builtin_amdgcn_wmma_f32_16x16x4_f32(false, a, false, b,
                                                    (short)0, acc, false, false);
    }
    for (int r = 0; r < 8; ++r)
        ws[WS_T1 + (c0 + r + 8 * half) * 16 + l] = acc[r];
}

// T2[s,p] = sum_c x0[c,s]*ch2[c,p];  one wave per 16-row tile of s.
__global__ void k_t2(const float* __restrict__ data, const float* __restrict__ ch2,
                     float* ws) {
    const float mean = ws[0], sc = ws[1];
    const int lane = threadIdx.x, half = lane >> 4, l = lane & 15;
    const int s0 = blockIdx.x * 16;
    const int lc = (l < 12) ? l : 0;
    v8f acc = {};
    for (int c = 0; c < 256; c += 4) {
        const int k0 = c + 2 * half;
        v2f a, b;
        a.x = (data[k0 * 1024 + s0 + l]       - mean) * sc;
        a.y = (data[(k0 + 1) * 1024 + s0 + l] - mean) * sc;
        float bx = ch2[k0 * 12 + lc];
        float by = ch2[(k0 + 1) * 12 + lc];
        b.x = (l < 12) ? bx : 0.f;
        b.y = (l < 12) ? by : 0.f;
        acc = __builtin_amdgcn_wmma_f32_16x16x4_f32(false, a, false, b,
                                                    (short)0, acc, false, false);
    }
    for (int r = 0; r < 8; ++r)
        ws[WS_T2 + (s0 + r + 8 * half) * 16 + l] = acc[r];
}

// W = T1^T*ch1 + T2^T*sp2 (16x16, valid 12x12), then softmax rowsums -> LUTs.
__global__ void k_wmat(const float* __restrict__ ch1, const float* __restrict__ sp2,
                       const float* __restrict__ theta, const float* __restrict__ vel,
                       float* ws) {
    __shared__ float Wsh[256];
    const int lane = threadIdx.x, half = lane >> 4, l = lane & 15;
    const int lc = (l < 12) ? l : 0;
    v8f acc = {};
    // W1: A[p,k=c] = T1[c,p]  (K=256)
    for (int c = 0; c < 256; c += 4) {
        const int k0 = c + 2 * half;
        v2f a, b;
        a.x = ws[WS_T1 + k0 * 16 + l];
        a.y = ws[WS_T1 + (k0 + 1) * 16 + l];
        float bx = ch1[k0 * 12 + lc];
        float by = ch1[(k0 + 1) * 12 + lc];
        b.x = (l < 12) ? bx : 0.f;
        b.y = (l < 12) ? by : 0.f;
        acc = __builtin_amdgcn_wmma_f32_16x16x4_f32(false, a, false, b,
                                                    (short)0, acc, false, false);
    }
    // + W2: A[p,k=s] = T2[s,p]  (K=1024), same accumulator
    for (int s = 0; s < 1024; s += 4) {
        const int k0 = s + 2 * half;
        v2f a, b;
        a.x = ws[WS_T2 + k0 * 16 + l];
        a.y = ws[WS_T2 + (k0 + 1) * 16 + l];
        float bx = sp2[k0 * 12 + lc];
        float by = sp2[(k0 + 1) * 12 + lc];
        b.x = (l < 12) ? bx : 0.f;
        b.y = (l < 12) ? by : 0.f;
        acc = __builtin_amdgcn_wmma_f32_16x16x4_f32(false, a, false, b,
                                                    (short)0, acc, false, false);
    }
    for (int r = 0; r < 8; ++r)
        Wsh[(r + 8 * half) * 16 + l] = acc[r];
    __syncthreads();
    if (lane < 12) {
        float row[12], m = -3.0e38f;
        for (int q = 0; q < 12; ++q) { row[q] = Wsh[lane * 16 + q]; m = fmaxf(m, row[q]); }
        float S = 0.f;
        for (int q = 0; q < 12; ++q) S += __expf(row[q] - m);
        float rs = 0.f;
        for (int q = 0; q < 12; ++q) rs += __expf(row[q] - m) / S;  // == softmax rowsum
        ws[WS_THMOD + lane] = theta[lane] * rs;
        ws[WS_VMOD  + lane] = vel[lane]   * rs;
    }
}

// ---------------- pass 3: fused element-wise transform + output stats -------
__global__ void k_transform(const float* __restrict__ data, float* __restrict__ out,
                            long long n, float* ws) {
    __shared__ float thm[16], vm[16];
    __shared__ float ssum[TPB_RED], ssq[TPB_RED];
    if (threadIdx.x < 16) {
        thm[threadIdx.x] = ws[WS_THMOD + threadIdx.x];
        vm [threadIdx.x] = ws[WS_VMOD  + threadIdx.x];
    }
    __syncthreads();
    const float mean = ws[0], sc = ws[1];
    float s = 0.f, q = 0.f;
    long long stride = (long long)gridDim.x * blockDim.x;
    for (long long i = (long long)blockIdx.x * blockDim.x + threadIdx.x; i < n; i += stride) {
        float x = (data[i] - mean) * sc;
        // theta interpolation at idx = sigmoid(x)*11
        float idx = 11.f / (1.f + __expf(-x));
        int   p0  = (int)floorf(idx);
        float pos = idx - (float)p0;
        int p0c = p0 < 0 ? 0 : (p0 > 11 ? 11 : p0);
        int p1c = (p0 + 1) > 11 ? 11 : (p0 + 1 < 0 ? 0 : p0 + 1);
        float thv = (1.f - pos) * thm[p0c] + pos * thm[p1c];
        // velocity interpolation at idx2 = |tanh(x)|*11
        float idx2 = fabsf(tanhf(x)) * 11.f;
        int   q0   = (int)floorf(idx2);
        float pos2 = idx2 - (float)q0;
        int q0c = q0 > 11 ? 11 : q0;
        int q1c = (q0 + 1) > 11 ? 11 : q0 + 1;
        float dsv = (1.f - pos2) * vm[q0c] + pos2 * vm[q1c];
        float o = x * __expf(dsv * __sinf(thv)) + dsv * __cosf(thv);
        out[i] = o;
        s += o; q += o * o;
    }
    ssum[threadIdx.x] = s; ssq[threadIdx.x] = q;
    __syncthreads();
    for (int o = TPB_RED >> 1; o > 0; o >>= 1) {
        if ((int)threadIdx.x < o) {
            ssum[threadIdx.x] += ssum[threadIdx.x + o];
            ssq [threadIdx.x] += ssq [threadIdx.x + o];
        }
        __syncthreads();
    }
    if (threadIdx.x == 0) {
        ws[WS_P2 + 2 * blockIdx.x]     = ssum[0];
        ws[WS_P2 + 2 * blockIdx.x + 1] = ssq[0];
    }
}

__global__ void k_stats2b(float* ws, long long n, const float* __restrict__ oscale) {
    __shared__ float ssum[NBLK_RED], ssq[NBLK_RED];
    int t = threadIdx.x;
    ssum[t] = ws[WS_P2 + 2 * t];
    ssq [t] = ws[WS_P2 + 2 * t + 1];
    __syncthreads();
    for (int o = NBLK_RED >> 1; o > 0; o >>= 1) {
        if (t < o) { ssum[t] += ssum[t + o]; ssq[t] += ssq[t + o]; }
        __syncthreads();
    }
    if (t == 0) {
        double dn = (double)n;
        double m  = (double)ssum[0] / dn;
        double var = ((double)ssq[0] - dn * m * m) / (dn - 1.0);
        float inv = (float)(1.0 / sqrt(var));
        ws[2] = (float)m;
        ws[3] = inv * oscale[0];   // fold oscale
    }
}

// ---------------- pass 4: output normalization ------------------------------
__global__ void k_norm(float* __restrict__ out, long long n, const float* __restrict__ ws) {
    const float m = ws[2], s = ws[3];
    long long stride = (long long)gridDim.x * blockDim.x;
    for (long long i = (long long)blockIdx.x * blockDim.x + threadIdx.x; i < n; i += stride)
        out[i] = (out[i] - m) * s;
}

extern "C" void kernel_launch(void* const* d_in, const int* in_sizes, int n_in,
                              void* d_out, int out_size, void* d_ws, size_t ws_size,
                              hipStream_t stream) {
    const float* data   = (const float*)d_in[0];
    const float* iscale = (const float*)d_in[1];
    const float* oscale = (const float*)d_in[2];
    const float* ch1    = (const float*)d_in[3];
    const float* sp1    = (const float*)d_in[4];
    const float* ch2    = (const float*)d_in[5];
    const float* sp2    = (const float*)d_in[6];
    const float* theta  = (const float*)d_in[7];
    const float* vel    = (const float*)d_in[8];
    float* out = (float*)d_out;
    float* ws  = (float*)d_ws;
    long long n = (long long)in_sizes[0];   // B*C*H*W = 16,777,216

    k_stats1   <<<NBLK_RED, TPB_RED, 0, stream>>>(data, n, ws);
    k_stats1b  <<<1, NBLK_RED, 0, stream>>>(ws, n, iscale);
    k_t1       <<<16, 32, 0, stream>>>(data, sp1, ws);
    k_t2       <<<64, 32, 0, stream>>>(data, ch2, ws);
    k_wmat     <<<1, 32, 0, stream>>>(ch1, sp2, theta, vel, ws);
    k_transform<<<NBLK_RED, TPB_RED, 0, stream>>>(data, out, n, ws);
    k_stats2b  <<<1, NBLK_RED, 0, stream>>>(ws, n, oscale);
    k_norm     <<<NBLK_RED, TPB_RED, 0, stream>>>(out, n, ws);
}